// TaskAlignedAssiger_75557064671775
// MI455X (gfx1250) — compile-verified
//
#include <hip/hip_runtime.h>
#include <hip/hip_bf16.h>
#include <stdint.h>

#define TOPK 10
#define MAXM 32
#define MAX_A 8400
#define CHUNK 64
#define EPS_ASSIGN 1e-9f
#define EPS_CIOU   1e-7f

#define AS1 __attribute__((address_space(1)))
#define AS3 __attribute__((address_space(3)))

#if defined(__AMDGCN__) && defined(__has_builtin)
#if __has_builtin(__builtin_amdgcn_global_load_async_to_lds_b128) && __has_builtin(__builtin_amdgcn_s_wait_asynccnt)
#define ASYNC_LDS 1
#endif
#endif

#ifdef ASYNC_LDS
// Matches the builtin's parameter pointee type exactly (per compiler diagnostic):
// 'int __attribute__((__vector_size__(4 * sizeof(int)))) __device__ *'
typedef int v4i_b128 __attribute__((vector_size(4 * sizeof(int))));
#define ASYNC_B128(gsrc, ldst)                                                   \
    __builtin_amdgcn_global_load_async_to_lds_b128(                              \
        (AS1 v4i_b128*)(uintptr_t)(gsrc), (AS3 v4i_b128*)(uintptr_t)(ldst), 0, 0)
#endif

// CIoU between gt (g*) and pred (p*), clipped at 0, matching the reference's
// _ciou() exactly (eps only on heights, rho2/c2 penalty, arctan aspect term).
__device__ __forceinline__ float ciou_clipped(float gx1, float gy1, float gx2, float gy2,
                                              float px1, float py1, float px2, float py2)
{
    const float eps = EPS_CIOU;
    float w1 = gx2 - gx1;
    float h1 = gy2 - gy1 + eps;
    float w2 = px2 - px1;
    float h2 = py2 - py1 + eps;
    float iw = fmaxf(fminf(gx2, px2) - fmaxf(gx1, px1), 0.0f);
    float ih = fmaxf(fminf(gy2, py2) - fmaxf(gy1, py1), 0.0f);
    float inter = iw * ih;
    float uni   = w1 * h1 + w2 * h2 - inter + eps;
    float iou   = inter / uni;
    float cw = fmaxf(gx2, px2) - fminf(gx1, px1);
    float ch = fmaxf(gy2, py2) - fminf(gy1, py1);
    float c2 = cw * cw + ch * ch + eps;
    float dx = (px1 + px2) - (gx1 + gx2);
    float dy = (py1 + py2) - (gy1 + gy2);
    float rho2 = (dx * dx + dy * dy) * 0.25f;
    float dat = atanf(w2 / h2) - atanf(w1 / h1);
    const float pi2 = 3.14159265358979323846f * 3.14159265358979323846f;
    float v = (4.0f / pi2) * dat * dat;
    float al = v / (v - iou + (1.0f + eps));
    float c = iou - (rho2 / c2 + v * al);
    return fmaxf(c, 0.0f);
}

// overlap (= clip(ciou)*valid) and valid for gt m vs one anchor/pred box
__device__ __forceinline__ float ov_valid(float gx1, float gy1, float gx2, float gy2,
                                          float mask, float ax, float ay,
                                          float4 pb, float* validOut)
{
    float dmin = fminf(fminf(ax - gx1, ay - gy1), fminf(gx2 - ax, gy2 - ay));
    float valid = (dmin > EPS_ASSIGN) ? mask : 0.0f;
    *validOut = valid;
    if (valid == 0.0f) return 0.0f;
    return ciou_clipped(gx1, gy1, gx2, gy2, pb.x, pb.y, pb.z, pb.w) * valid;
}

// ---------------------------------------------------------------------------
// K1: one block per (b,m) row. Compute align row into LDS, select top-10 with
// jax top_k tie-breaking (value desc, index asc), store indices pre-filtered
// by in_gts && mask. Also zero-init gmm/gmo accumulators for this row.
// ---------------------------------------------------------------------------
__global__ void __launch_bounds__(256) k1_topk(
    const float* __restrict__ pd_scores, const float* __restrict__ pd_bboxes,
    const float* __restrict__ anc, const int* __restrict__ gt_labels,
    const float* __restrict__ gt_bboxes, const float* __restrict__ gt_mask,
    int A, int M, int C,
    int* __restrict__ topk_out, unsigned* __restrict__ gmm, unsigned* __restrict__ gmo)
{
    __shared__ float s_align[MAX_A];
    __shared__ unsigned long long s_key;
    __shared__ int s_sel[TOPK];

    int row = blockIdx.x;          // b*M + m
    int b = row / M;
    float gx1 = gt_bboxes[row * 4 + 0], gy1 = gt_bboxes[row * 4 + 1];
    float gx2 = gt_bboxes[row * 4 + 2], gy2 = gt_bboxes[row * 4 + 3];
    float mask = gt_mask[row];
    int   lab  = gt_labels[row];
    if (threadIdx.x == 0) { gmm[row] = 0u; gmo[row] = 0u; }

    const float* sc = pd_scores + (size_t)b * A * C;
    for (int a = threadIdx.x; a < A; a += blockDim.x) {
        float2 ap = ((const float2*)anc)[a];
        float al = 0.0f;
        float dmin = fminf(fminf(ap.x - gx1, ap.y - gy1), fminf(gx2 - ap.x, gy2 - ap.y));
        float valid = (dmin > EPS_ASSIGN) ? mask : 0.0f;
        if (valid != 0.0f) {
            const float* sp = sc + (size_t)a * C + lab;
            __builtin_prefetch(sp, 0, 1);     // global_prefetch_b8: hide the gather
            float4 pb = ((const float4*)pd_bboxes)[(size_t)b * A + a];
            float ov = ciou_clipped(gx1, gy1, gx2, gy2, pb.x, pb.y, pb.z, pb.w) * valid;
            float p2 = ov * ov;
            al = (*sp) * (p2 * p2 * p2) * valid;   // score^1 * ov^6 * valid
        }
        s_align[a] = al;
    }
    __syncthreads();

    // 10 rounds of block argmax over LDS. Key packs (value_bits<<32)|(~idx)
    // (all values >= 0 so float bits are monotonic); ties -> lowest index,
    // exactly like jax.lax.top_k. Selected entries set to -1 (excluded).
    for (int k = 0; k < TOPK; ++k) {
        if (threadIdx.x == 0) s_key = 0ull;
        __syncthreads();
        unsigned long long best = 0ull;
        for (int a = threadIdx.x; a < A; a += blockDim.x) {
            float v = s_align[a];
            if (v < 0.0f) continue;
            unsigned long long key = ((unsigned long long)__float_as_uint(v) << 32)
                                   | (unsigned long long)(0xFFFFFFFFu - (unsigned)a);
            best = (key > best) ? key : best;
        }
        atomicMax(&s_key, best);
        __syncthreads();
        if (threadIdx.x == 0) {
            int idx = (int)(0xFFFFFFFFu - (unsigned)(s_key & 0xFFFFFFFFull));
            s_sel[k] = idx;
            s_align[idx] = -1.0f;
        }
        __syncthreads();
    }

    if (threadIdx.x < TOPK) {
        int aidx = s_sel[threadIdx.x];
        float2 ap = ((const float2*)anc)[aidx];
        float dmin = fminf(fminf(ap.x - gx1, ap.y - gy1), fminf(gx2 - ap.x, gy2 - ap.y));
        bool ok = (dmin > EPS_ASSIGN) && (mask != 0.0f);   // pos_mask0 membership
        topk_out[row * TOPK + threadIdx.x] = ok ? aidx : -1;
    }
}

// ---------------------------------------------------------------------------
// K2: one thread per (b,a). Build pos0 bitmask, resolve fg>1 via overlaps
// argmax (recomputed CIoU; first-max tie rule), emit t_bboxes + fg, feed
// gmm/gmo via monotonic uint atomicMax, store msel/alignsel for K4.
// GT tables staged into LDS via gfx1250 async global->LDS DMA when available.
// ---------------------------------------------------------------------------
__global__ void __launch_bounds__(256) k2_assign(
    const float* __restrict__ pd_scores, const float* __restrict__ pd_bboxes,
    const float* __restrict__ anc, const int* __restrict__ gt_labels,
    const float* __restrict__ gt_bboxes, const float* __restrict__ gt_mask,
    const int* __restrict__ topk, int A, int M, int C,
    float* __restrict__ out_tb, float* __restrict__ out_fg,
    int* __restrict__ msel, float* __restrict__ aselw,
    unsigned* __restrict__ gmm, unsigned* __restrict__ gmo)
{
    __shared__ __align__(16) float s_box[MAXM * 4];
    __shared__ __align__(16) int   s_lab[MAXM];
    __shared__ __align__(16) float s_msk[MAXM];
    __shared__ __align__(16) int   s_top[MAXM * TOPK];

    int b = blockIdx.y;
    int t = threadIdx.x;

#ifdef ASYNC_LDS
    if (M == MAXM) {
        const float* gb = gt_bboxes + (size_t)b * M * 4;
        const int*   gl = gt_labels + (size_t)b * M;
        const float* gm = gt_mask   + (size_t)b * M;
        const int*   tk = topk      + (size_t)b * M * TOPK;
        if (t < 32) {
            ASYNC_B128(gb + t * 4, &s_box[t * 4]);
        } else if (t < 40) {
            int j = t - 32;
            ASYNC_B128(gl + j * 4, &s_lab[j * 4]);
        } else if (t < 48) {
            int j = t - 40;
            ASYNC_B128(gm + j * 4, &s_msk[j * 4]);
        } else if (t < 128) {
            int j = t - 48;
            ASYNC_B128(tk + j * 4, &s_top[j * 4]);
        }
        __builtin_amdgcn_s_wait_asynccnt(0);
    } else
#endif
    {
        for (int i = t; i < M * 4; i += blockDim.x) s_box[i] = gt_bboxes[(size_t)b * M * 4 + i];
        for (int i = t; i < M; i += blockDim.x) {
            s_lab[i] = gt_labels[(size_t)b * M + i];
            s_msk[i] = gt_mask[(size_t)b * M + i];
        }
        for (int i = t; i < M * TOPK; i += blockDim.x) s_top[i] = topk[(size_t)b * M * TOPK + i];
    }
    __syncthreads();

    int a = blockIdx.x * blockDim.x + t;
    if (a >= A) return;

    float2 ap = ((const float2*)anc)[a];
    float4 pb = ((const float4*)pd_bboxes)[(size_t)b * A + a];

    unsigned pos0 = 0u;
    for (int m = 0; m < M; ++m) {
        const int* tp = &s_top[m * TOPK];
        bool mem = false;
#pragma unroll
        for (int j = 0; j < TOPK; ++j) mem = mem || (tp[j] == a);
        pos0 |= (mem ? (1u << m) : 0u);
    }
    int fg0 = __popc(pos0);
    int ms = -1;
    if (fg0 == 1) {
        ms = __ffs(pos0) - 1;
    } else if (fg0 > 1) {
        // argmax over overlaps column, first-max tie rule (jnp.argmax).
        float bestv = -1.0f;
        for (int m = 0; m < M; ++m) {
            float vtmp;
            float ov = ov_valid(s_box[m * 4], s_box[m * 4 + 1], s_box[m * 4 + 2], s_box[m * 4 + 3],
                                s_msk[m], ap.x, ap.y, pb, &vtmp);
            if (ov > bestv) { bestv = ov; ms = m; }
        }
    }

    int tgt = (ms >= 0) ? ms : 0;   // argmax of all-zero column -> 0, like jnp
    float4 tb = make_float4(s_box[tgt * 4], s_box[tgt * 4 + 1], s_box[tgt * 4 + 2], s_box[tgt * 4 + 3]);
    ((float4*)out_tb)[(size_t)b * A + a] = tb;
    out_fg[(size_t)b * A + a] = (ms >= 0) ? 1.0f : 0.0f;

    float asel = 0.0f;
    if (ms >= 0) {
        float valid;
        float ov = ov_valid(s_box[ms * 4], s_box[ms * 4 + 1], s_box[ms * 4 + 2], s_box[ms * 4 + 3],
                            s_msk[ms], ap.x, ap.y, pb, &valid);
        float score = pd_scores[((size_t)b * A + a) * C + s_lab[ms]];
        float p2 = ov * ov;
        asel = score * (p2 * p2 * p2) * valid;
        // values >= 0: float bit pattern is monotonic under unsigned compare
        atomicMax(&gmm[b * M + ms], __float_as_uint(asel));
        atomicMax(&gmo[b * M + ms], __float_as_uint(ov));
    }
    msel[(size_t)b * A + a] = ms;
    aselw[(size_t)b * A + a] = asel;
}

// ---------------------------------------------------------------------------
// K4: t_labels. Per 64-anchor tile: 64 threads compute norm/label into LDS,
// then all 256 threads stream the 80-wide one-hot rows with coalesced stores.
// ---------------------------------------------------------------------------
__global__ void __launch_bounds__(256) k4_labels(
    const int* __restrict__ gt_labels, const int* __restrict__ msel,
    const float* __restrict__ aselw, const unsigned* __restrict__ gmm,
    const unsigned* __restrict__ gmo, int A, int M, int C,
    float* __restrict__ out_tl)
{
    __shared__ float s_norm[CHUNK];
    __shared__ int   s_lab[CHUNK];
    int b  = blockIdx.y;
    int a0 = blockIdx.x * CHUNK;
    int t  = threadIdx.x;

    if (t < CHUNK) {
        int a = a0 + t;
        float nrm = 0.0f;
        int lb = -1;
        if (a < A) {
            int ms = msel[(size_t)b * A + a];
            if (ms >= 0) {
                float gm = __uint_as_float(gmm[b * M + ms]);
                float go = __uint_as_float(gmo[b * M + ms]);
                nrm = aselw[(size_t)b * A + a] * go / (gm + EPS_ASSIGN);
                lb = gt_labels[(size_t)b * M + ms];
            }
        }
        s_norm[t] = nrm;
        s_lab[t]  = lb;
    }
    __syncthreads();

    int nA = min(CHUNK, A - a0);
    if (nA <= 0) return;
    int limit = nA * C;
    float* dst = out_tl + ((size_t)b * A + a0) * (size_t)C;
    for (int i = t; i < limit; i += blockDim.x) {
        int al = i / C;
        int c  = i - al * C;
        dst[i] = (c == s_lab[al]) ? s_norm[al] : 0.0f;
    }
}

extern "C" void kernel_launch(void* const* d_in, const int* in_sizes, int n_in,
                              void* d_out, int out_size, void* d_ws, size_t ws_size,
                              hipStream_t stream) {
    const float* pd_scores = (const float*)d_in[0];
    const float* pd_bboxes = (const float*)d_in[1];
    const float* anc       = (const float*)d_in[2];
    const int*   gt_labels = (const int*)d_in[3];
    const float* gt_bboxes = (const float*)d_in[4];
    const float* gt_mask   = (const float*)d_in[5];

    int A  = in_sizes[2] / 2;                 // 8400
    int bs = in_sizes[1] / (A * 4);           // 16
    int M  = in_sizes[4] / (bs * 4);          // 32
    int C  = in_sizes[0] / (bs * A);          // 80

    float* out    = (float*)d_out;
    float* out_tb = out;                                   // (bs, A, 4)
    float* out_tl = out + (size_t)bs * A * 4;              // (bs, A, C)
    float* out_fg = out_tl + (size_t)bs * A * C;           // (bs, A)

    char* ws = (char*)d_ws;
    auto alignup = [](size_t x) { return (x + 255) & ~(size_t)255; };
    size_t off = 0;
    int*      topk = (int*)(ws + off);      off = alignup(off + (size_t)bs * M * TOPK * sizeof(int));
    unsigned* gmm  = (unsigned*)(ws + off); off = alignup(off + (size_t)bs * M * sizeof(unsigned));
    unsigned* gmo  = (unsigned*)(ws + off); off = alignup(off + (size_t)bs * M * sizeof(unsigned));
    int*      msel = (int*)(ws + off);      off = alignup(off + (size_t)bs * A * sizeof(int));
    float*    asel = (float*)(ws + off);    off = alignup(off + (size_t)bs * A * sizeof(float));
    (void)n_in; (void)out_size; (void)ws_size;

    k1_topk<<<bs * M, 256, 0, stream>>>(pd_scores, pd_bboxes, anc, gt_labels, gt_bboxes,
                                        gt_mask, A, M, C, topk, gmm, gmo);

    dim3 g2((A + 255) / 256, bs);
    k2_assign<<<g2, 256, 0, stream>>>(pd_scores, pd_bboxes, anc, gt_labels, gt_bboxes,
                                      gt_mask, topk, A, M, C, out_tb, out_fg,
                                      msel, asel, gmm, gmo);

    dim3 g4((A + CHUNK - 1) / CHUNK, bs);
    k4_labels<<<g4, 256, 0, stream>>>(gt_labels, msel, asel, gmm, gmo, A, M, C, out_tl);
}